// ViewTransform_22411139351165
// MI455X (gfx1250) — compile-verified
//
#include <hip/hip_runtime.h>

// Problem constants (match reference setup_inputs()).
#define BEV_H   256
#define BEV_W   128
#define NBEV    (BEV_H * BEV_W)          // 32768
#define B_      4
#define N_      6
#define C_      128
#define H_      64
#define W_      176
#define HW_     (H_ * W_)                // 11264
#define CCHUNK  16                       // channels per block (gridDim.y = C_/CCHUNK)
#define CSTAGE  8                        // channels per async pipeline stage
#define PIXBLK  256                      // pixels per block (8 waves * 32)

#if __has_builtin(__builtin_amdgcn_s_wait_asynccnt)
#define WAIT_ASYNC(n) __builtin_amdgcn_s_wait_asynccnt(n)
#else
#define WAIT_ASYNC(n) asm volatile("s_wait_asynccnt %0" ::"i"(n))
#endif

// ---------------------------------------------------------------------------
// Zero-fill the BEV output (harness poisons d_out; atomics need zeroed base).
// Default RT stores leave the 67 MB output resident in the 192 MB L2, where
// the subsequent atomic accumulation resolves.
// ---------------------------------------------------------------------------
__global__ __launch_bounds__(256) void bev_zero(float4* __restrict__ out) {
    size_t i = (size_t)blockIdx.x * 256 + threadIdx.x;
    out[i] = make_float4(0.f, 0.f, 0.f, 0.f);
}

// ---------------------------------------------------------------------------
// Async-DMA one pipeline stage: CSTAGE channel rows of PIXBLK floats (8 KB)
// from global (channel stride HW_) into an LDS buffer, via
// GLOBAL_LOAD_ASYNC_TO_LDS_B128 (16 B per lane, coalesced, tracked by
// ASYNCcnt). Feature data is read exactly once, so it is loaded with
// th:TH_LOAD_NT to avoid evicting the L2-resident output accumulation lines.
// 8 KB = 512 x 16B chunks; 256 threads -> 2 async issues per thread.
// ---------------------------------------------------------------------------
__device__ __forceinline__ void stage_async(const float* __restrict__ gbase,
                                            float (*buf)[PIXBLK], int tid) {
    #pragma unroll
    for (int i = 0; i < 2; ++i) {
        const int chunk = i * 256 + tid;           // 0..511
        const int ch    = chunk >> 6;              // 64 x 16B chunks per 1KB row
        const int q     = chunk & 63;
        const float* g  = gbase + (size_t)ch * HW_ + (q << 2);
        const unsigned lds = (unsigned)(unsigned long long)
            (__attribute__((address_space(3))) float*)&buf[ch][q << 2];
        asm volatile("global_load_async_to_lds_b128 %0, %1, off th:TH_LOAD_NT"
                     :: "v"(lds), "v"((unsigned long long)g)
                     : "memory");
    }
}

// ---------------------------------------------------------------------------
// Fused geometry + voxel-pool scatter with a 2-deep ASYNCcnt pipeline.
//
// Block = 256 threads = 8 wave32; each block owns 256 consecutive pixels of
// one (b,n) image (HW_ % 256 == 0 -> b,n,K,E wave-uniform -> SGPRs).
// gridDim.y picks a 16-channel chunk; geometry is computed once per thread.
//
// Consecutive pixels overwhelmingly share a BEV cell (a whole image spans only
// a few cells: camx = (x-cx)/fx spans ~±0.7 m ~ 3 cells), so each wave does a
// segmented inclusive scan over its 32 lanes and only segment-tail lanes issue
// one relaxed agent-scope global_atomic_add_f32 (accumulation stays in L2:
// output is 67 MB < 192 MB L2). This cuts atomics ~15-30x vs naive scatter.
// ---------------------------------------------------------------------------
__global__ __launch_bounds__(256) void bev_scatter(
    const float* __restrict__ feats,   // (B,N,C,H,W)
    const float* __restrict__ intr,    // (B,N,3,3)
    const float* __restrict__ extr,    // (B,N,4,4)
    float* __restrict__ out)           // (B,C,BEV_H,BEV_W)
{
    __shared__ float tile[2][CSTAGE][PIXBLK];      // 16 KB double buffer

    const int BLOCKS_PER_IMG = HW_ / PIXBLK;       // 44
    const int img      = blockIdx.x / BLOCKS_PER_IMG;        // b*N_+n (uniform)
    const int pixStart = (blockIdx.x % BLOCKS_PER_IMG) * PIXBLK;
    const int tid      = threadIdx.x;
    const int pix      = pixStart + tid;
    const int b        = img / N_;

    // ---- kick off both DMA stages first (overlap with geometry ALU) --------
    const float* gbase = feats + ((size_t)img * C_ + (size_t)blockIdx.y * CCHUNK) * HW_ + pixStart;
    stage_async(gbase,                        tile[0], tid);
    stage_async(gbase + (size_t)CSTAGE * HW_, tile[1], tid);

    // ---- geometry: pixel -> BEV cell (analytic pinhole inverse) ------------
    const float* K = intr + (size_t)img * 9;       // wave-uniform loads
    const float* E = extr + (size_t)img * 16;
    const float fx = K[0], cx = K[2], fy = K[4], cy = K[5];

    const float px = (float)(pix % W_);
    const float py = (float)(pix / W_);
    const float camx = (px - cx) / fx;
    const float camy = (py - cy) / fy;
    // world = E * [camx, camy, 1, 1]  (rows 0 and 1; z unused)
    const float wx = E[0] * camx + E[1] * camy + E[2] + E[3];
    const float wy = E[4] * camx + E[5] * camy + E[6] + E[7];

    // trunc-toward-zero matches jnp.trunc(...).astype(int32); /0.5 == *2
    const int bx = (int)truncf(wx * 2.0f) + (BEV_W / 2);
    const int by = (int)truncf(wy * 2.0f) + (BEV_H / 2);
    const bool valid = (bx >= 0) & (bx < BEV_W) & (by >= 0) & (by < BEV_H);
    const int  idx   = valid ? (by * BEV_W + bx) : -1;

    // ---- wave32 segment structure (computed once, reused for all channels) -
    const int lane = tid & 31;
    const int prevIdx = __shfl_up(idx, 1, 32);
    const bool head = (lane == 0) || (idx != prevIdx);
    const unsigned hm = (unsigned)__ballot(head);  // wave32: low 32 bits
    const unsigned le = hm & (0xFFFFFFFFu >> (31 - lane));
    const int  hp   = 31 - __builtin_clz(le);      // my segment's head lane
    const bool tail = (lane == 31) || ((hm >> (lane + 1)) & 1u);
    const bool doAtomic = tail && (idx >= 0);

    float* op = out + ((size_t)b * C_ + (size_t)blockIdx.y * CCHUNK) * (size_t)NBEV;

    // ---- pipeline: wait stage0 (stage1 DMA still in flight), compute -------
    #pragma unroll
    for (int s = 0; s < 2; ++s) {
        if (s == 0) WAIT_ASYNC(2);                 // first 2 of my 4 async ops done
        else        WAIT_ASYNC(0);
        __syncthreads();                           // whole tile[s] valid

        #pragma unroll 4
        for (int c = 0; c < CSTAGE; ++c) {
            float v = tile[s][c][tid];             // conflict-free ds_load_b32

            // segmented inclusive scan across the wave
            #pragma unroll
            for (int d = 1; d < 32; d <<= 1) {
                float u = __shfl_up(v, d, 32);
                if (lane - d >= hp) v += u;
            }

            if (doAtomic) {
                // non-returning global_atomic_add_f32, L2-resident accumulation
                (void)__hip_atomic_fetch_add(
                    op + (size_t)(s * CSTAGE + c) * NBEV + idx, v,
                    __ATOMIC_RELAXED, __HIP_MEMORY_SCOPE_AGENT);
            }
        }
    }
}

// ---------------------------------------------------------------------------
extern "C" void kernel_launch(void* const* d_in, const int* in_sizes, int n_in,
                              void* d_out, int out_size, void* d_ws, size_t ws_size,
                              hipStream_t stream) {
    (void)in_sizes; (void)n_in; (void)d_ws; (void)ws_size; (void)out_size;

    const float* feats = (const float*)d_in[0];   // (B,N,C,H,W) f32
    const float* intr  = (const float*)d_in[1];   // (B,N,3,3)  f32
    const float* extr  = (const float*)d_in[2];   // (B,N,4,4)  f32
    float*       out   = (float*)d_out;           // (B,C,256,128) f32

    // 1) zero the accumulation target
    const int n4 = (B_ * C_ * NBEV) / 4;          // 4,194,304
    bev_zero<<<n4 / 256, 256, 0, stream>>>((float4*)out);

    // 2) fused geometry + async-pipelined segmented scatter-add
    dim3 grid(B_ * N_ * (HW_ / PIXBLK), C_ / CCHUNK);  // (1056, 8)
    bev_scatter<<<grid, PIXBLK, 0, stream>>>(feats, intr, extr, out);
}